// GNNStructEncoder_52716428591752
// MI455X (gfx1250) — compile-verified
//
#include <hip/hip_runtime.h>
#include <hip/hip_bf16.h>

#define NN 50000
#define DD 128
#define EE 800000

typedef __attribute__((ext_vector_type(2))) float v2f;
typedef __attribute__((ext_vector_type(8))) float v8f;

// ---------------------------------------------------------------------------
// WMMA f32 GEMM:  C[n,O] = act( A[n,K] * op(W) + bias )
//   TRANSB=1 : op(W) = W^T with W stored [O,K] row-major  (torch Linear)
//   TRANSB=0 : op(W) = W   with W stored [K,O] row-major  (res_W)
// One wave32 computes a 16x64 strip: 4 accumulators, A fragment loaded once
// per k-step and reused by 4 V_WMMA_F32_16X16X4_F32 ops (4 wmma : 5 loads).
// Fragment layouts (CDNA5 ISA 7.12.2):
//   A 16x4 : lane&15 = M, half=lane>>4 : v0=A[M,2h], v1=A[M,2h+1]
//   B 4x16 : lane&15 = N               : v0=B[2h,N], v1=B[2h+1,N]
//   C 16x16: vgpr v : row = v + 8*half, col = lane&15
// ---------------------------------------------------------------------------
template <bool TRANSB, int ACT>
__global__ __launch_bounds__(256) void gemm_wmma(
    const float* __restrict__ A, const float* __restrict__ W,
    const float* __restrict__ bias, float* __restrict__ C,
    int n, int K, int O)
{
    const int wave = threadIdx.x >> 5;
    const int lane = threadIdx.x & 31;
    const int groups_n = O >> 6;               // 64-wide column groups
    const int total = (n >> 4) * groups_n;
    const int tile = blockIdx.x * 8 + wave;
    if (tile >= total) return;                 // wave-uniform: EXEC stays full

    const int tm = tile / groups_n;
    const int tg = tile % groups_n;
    const int half = lane >> 4;
    const int l15  = lane & 15;
    const int rbase = tm << 4;
    const int cbase = tg << 6;
    const int arow  = rbase + l15;             // A row this lane supplies

    v8f acc[4];
#pragma unroll
    for (int t = 0; t < 4; ++t) acc[t] = (v8f){0.f,0.f,0.f,0.f,0.f,0.f,0.f,0.f};

#if __has_builtin(__builtin_amdgcn_wmma_f32_16x16x4_f32)
    const float* Ap = A + (size_t)arow * K;
    for (int k0 = 0; k0 < K; k0 += 4) {
        const int ka = k0 + 2 * half;
        v2f a = *(const v2f*)(Ap + ka);
        v2f b[4];
        if (TRANSB) {
            const float* Wp = W + (size_t)(cbase + l15) * K + ka;
#pragma unroll
            for (int t = 0; t < 4; ++t)
                b[t] = *(const v2f*)(Wp + (size_t)(16 * t) * K);
        } else {
            const float* Wp = W + (size_t)ka * O + cbase + l15;
#pragma unroll
            for (int t = 0; t < 4; ++t) {
                b[t].x = Wp[16 * t];
                b[t].y = Wp[16 * t + O];
            }
        }
#pragma unroll
        for (int t = 0; t < 4; ++t)
            acc[t] = __builtin_amdgcn_wmma_f32_16x16x4_f32(
                /*neg_a=*/false, a, /*neg_b=*/false, b[t],
                /*c_mod=*/(short)0, acc[t], /*reuse_a=*/false, /*reuse_b=*/false);
    }
#else
    // scalar fallback (keeps compile alive if builtin signature differs)
    for (int k = 0; k < K; ++k) {
#pragma unroll
        for (int t = 0; t < 4; ++t) {
            int col = cbase + 16 * t + l15;
            float bv = TRANSB ? W[(size_t)col * K + k] : W[(size_t)k * O + col];
#pragma unroll
            for (int v = 0; v < 8; ++v)
                acc[t][v] += A[(size_t)(rbase + v + 8 * half) * K + k] * bv;
        }
    }
#endif

#pragma unroll
    for (int t = 0; t < 4; ++t) {
        const int col = cbase + 16 * t + l15;
        const float bv = bias ? bias[col] : 0.f;
#pragma unroll
        for (int v = 0; v < 8; ++v) {
            float val = acc[t][v] + bv;
            if (ACT == 1) val = fmaxf(val, 0.f);
            C[(size_t)(rbase + v + 8 * half) * O + col] = val;
        }
    }
}

// ---------------------------------------------------------------------------
// helpers
// ---------------------------------------------------------------------------
__device__ __forceinline__ float wave_sum32(float v) {
#pragma unroll
    for (int m = 16; m > 0; m >>= 1) v += __shfl_xor(v, m, 32);
    return v;
}

__global__ __launch_bounds__(256) void fill_kernel(float* __restrict__ p, float v, int n) {
    int i = blockIdx.x * blockDim.x + threadIdx.x;
    if (i < n) p[i] = v;
}

__global__ __launch_bounds__(256) void deg_scatter_kernel(const int* __restrict__ dst,
                                                          float* __restrict__ deg, int e) {
    int i = blockIdx.x * blockDim.x + threadIdx.x;
    if (i < e) atomicAdd(&deg[dst[i]], 1.0f);
}

__global__ __launch_bounds__(256) void rsqrt_kernel(float* __restrict__ p, int n) {
    int i = blockIdx.x * blockDim.x + threadIdx.x;
    if (i < n) p[i] = rsqrtf(p[i]);
}

// residual[dst] += v[src]   (one wave per edge, 4 dims per lane)
__global__ __launch_bounds__(256) void res_scatter_kernel(
    const int* __restrict__ src, const int* __restrict__ dst,
    const float* __restrict__ v, float* __restrict__ o, int e)
{
    int wv = (blockIdx.x * 256 + threadIdx.x) >> 5;
    int lane = threadIdx.x & 31;
    if (wv >= e) return;
    const float* vr = v + (size_t)src[wv] * DD;
    float* orow = o + (size_t)dst[wv] * DD;
#pragma unroll
    for (int j = 0; j < 4; ++j) {
        int d = lane + 32 * j;
        atomicAdd(&orow[d], vr[d]);
    }
}

// out[n,d] = hw[n,d]*isd[n]^2 + bias[d]   (elementwise over N*D)
__global__ __launch_bounds__(256) void gcn_base_kernel(
    const float* __restrict__ hw, const float* __restrict__ isd,
    const float* __restrict__ bias, float* __restrict__ o, int total)
{
    int i = blockIdx.x * blockDim.x + threadIdx.x;
    if (i >= total) return;
    int nd = i >> 7, d = i & 127;
    float s = isd[nd];
    o[i] = hw[i] * s * s + bias[d];
}

// out[dst] += hw[src] * isd[src]*isd[dst]
__global__ __launch_bounds__(256) void gcn_scatter_kernel(
    const int* __restrict__ src, const int* __restrict__ dst,
    const float* __restrict__ hw, const float* __restrict__ isd,
    float* __restrict__ o, int e)
{
    int wv = (blockIdx.x * 256 + threadIdx.x) >> 5;
    int lane = threadIdx.x & 31;
    if (wv >= e) return;
    int s = src[wv], t = dst[wv];
    float w = isd[s] * isd[t];
    const float* vr = hw + (size_t)s * DD;
    float* orow = o + (size_t)t * DD;
#pragma unroll
    for (int j = 0; j < 4; ++j) {
        int d = lane + 32 * j;
        atomicAdd(&orow[d], vr[d] * w);
    }
}

// in-place LayerNorm(128) + relu, one wave per node
__global__ __launch_bounds__(256) void ln128_relu_kernel(
    float* __restrict__ h, const float* __restrict__ g, const float* __restrict__ b, int n)
{
    int wv = (blockIdx.x * 256 + threadIdx.x) >> 5;
    int lane = threadIdx.x & 31;
    if (wv >= n) return;
    float* row = h + (size_t)wv * DD;
    float x[4], s = 0.f;
#pragma unroll
    for (int j = 0; j < 4; ++j) { x[j] = row[lane + 32 * j]; s += x[j]; }
    float mean = wave_sum32(s) * (1.f / 128.f);
    float sq = 0.f;
#pragma unroll
    for (int j = 0; j < 4; ++j) { float d = x[j] - mean; sq += d * d; }
    float inv = rsqrtf(wave_sum32(sq) * (1.f / 128.f) + 1e-5f);
#pragma unroll
    for (int j = 0; j < 4; ++j) {
        int d = lane + 32 * j;
        row[d] = fmaxf((x[j] - mean) * inv * g[d] + b[d], 0.f);
    }
}

// 2x2 single-head attention per node (seq len 2), one wave per node
__global__ __launch_bounds__(256) void attention_kernel(
    const float* __restrict__ qkv0, const float* __restrict__ qkv1,
    float* __restrict__ ao0, float* __restrict__ ao1, int n)
{
    int wv = (blockIdx.x * 256 + threadIdx.x) >> 5;
    int lane = threadIdx.x & 31;
    if (wv >= n) return;
    const float* b0 = qkv0 + (size_t)wv * 384;
    const float* b1 = qkv1 + (size_t)wv * 384;
    float q0[4], k0[4], v0[4], q1[4], k1[4], v1[4];
#pragma unroll
    for (int j = 0; j < 4; ++j) {
        int d = lane + 32 * j;
        q0[j] = b0[d]; k0[j] = b0[128 + d]; v0[j] = b0[256 + d];
        q1[j] = b1[d]; k1[j] = b1[128 + d]; v1[j] = b1[256 + d];
    }
    float l00 = 0, l01 = 0, l10 = 0, l11 = 0;
#pragma unroll
    for (int j = 0; j < 4; ++j) {
        l00 += q0[j] * k0[j]; l01 += q0[j] * k1[j];
        l10 += q1[j] * k0[j]; l11 += q1[j] * k1[j];
    }
    l00 = wave_sum32(l00); l01 = wave_sum32(l01);
    l10 = wave_sum32(l10); l11 = wave_sum32(l11);
    const float scale = 0.08838834764831845f;  // rsqrt(128)
    l00 *= scale; l01 *= scale; l10 *= scale; l11 *= scale;
    float m0 = fmaxf(l00, l01), e00 = __expf(l00 - m0), e01 = __expf(l01 - m0);
    float r0 = 1.f / (e00 + e01), w00 = e00 * r0, w01 = e01 * r0;
    float m1 = fmaxf(l10, l11), e10 = __expf(l10 - m1), e11 = __expf(l11 - m1);
    float r1 = 1.f / (e10 + e11), w10 = e10 * r1, w11 = e11 * r1;
    float* o0 = ao0 + (size_t)wv * DD;
    float* o1 = ao1 + (size_t)wv * DD;
#pragma unroll
    for (int j = 0; j < 4; ++j) {
        int d = lane + 32 * j;
        o0[d] = w00 * v0[j] + w01 * v1[j];
        o1[d] = w10 * v0[j] + w11 * v1[j];
    }
}

// fused = LayerNorm384(concat(ao0, ao1, residual)), one wave per node
__global__ __launch_bounds__(256) void fuse_ln384_kernel(
    const float* __restrict__ a0, const float* __restrict__ a1,
    const float* __restrict__ rs, const float* __restrict__ g,
    const float* __restrict__ b, float* __restrict__ o, int n)
{
    int wv = (blockIdx.x * 256 + threadIdx.x) >> 5;
    int lane = threadIdx.x & 31;
    if (wv >= n) return;
    const float* p0 = a0 + (size_t)wv * DD;
    const float* p1 = a1 + (size_t)wv * DD;
    const float* p2 = rs + (size_t)wv * DD;
    float x[12], s = 0.f;
#pragma unroll
    for (int j = 0; j < 12; ++j) {
        int d = lane + 32 * j;
        float v = (d < 128) ? p0[d] : (d < 256) ? p1[d - 128] : p2[d - 256];
        x[j] = v; s += v;
    }
    float mean = wave_sum32(s) * (1.f / 384.f);
    float sq = 0.f;
#pragma unroll
    for (int j = 0; j < 12; ++j) { float d = x[j] - mean; sq += d * d; }
    float inv = rsqrtf(wave_sum32(sq) * (1.f / 384.f) + 1e-5f);
    float* orow = o + (size_t)wv * 384;
#pragma unroll
    for (int j = 0; j < 12; ++j) {
        int d = lane + 32 * j;
        orow[d] = (x[j] - mean) * inv * g[d] + b[d];
    }
}

// column sums of pre-out (for PairNorm col_mean); 128 threads per block
__global__ void colsum_kernel(const float* __restrict__ p, float* __restrict__ cs, int n) {
    int c = threadIdx.x;
    float s = 0.f;
    for (int r = blockIdx.x; r < n; r += gridDim.x) s += p[(size_t)r * DD + c];
    atomicAdd(&cs[c], s);
}

// PairNorm PN-SCS: out = 20*pre/rownorm - colsum/N, one wave per node
__global__ __launch_bounds__(256) void pairnorm_kernel(
    const float* __restrict__ pre, const float* __restrict__ cs,
    float* __restrict__ o, int n)
{
    int wv = (blockIdx.x * 256 + threadIdx.x) >> 5;
    int lane = threadIdx.x & 31;
    if (wv >= n) return;
    const float* row = pre + (size_t)wv * DD;
    float x[4], sq = 0.f;
#pragma unroll
    for (int j = 0; j < 4; ++j) { x[j] = row[lane + 32 * j]; sq += x[j] * x[j]; }
    float rn = sqrtf(1e-6f + wave_sum32(sq));
    float inv = 20.f / rn;
    float* orow = o + (size_t)wv * DD;
    const float invN = 1.f / (float)NN;
#pragma unroll
    for (int j = 0; j < 4; ++j) {
        int d = lane + 32 * j;
        orow[d] = x[j] * inv - cs[d] * invN;
    }
}

// ---------------------------------------------------------------------------
extern "C" void kernel_launch(void* const* d_in, const int* in_sizes, int n_in,
                              void* d_out, int out_size, void* d_ws, size_t ws_size,
                              hipStream_t stream)
{
    const float* x      = (const float*)d_in[0];
    const float* x_org  = (const float*)d_in[1];
    const int*   ei     = (const int*)d_in[2];
    const float* in_W   = (const float*)d_in[3];
    const float* in_b   = (const float*)d_in[4];
    const float* res_W  = (const float*)d_in[5];
    const float* gcn_W  = (const float*)d_in[6];
    const float* gcn_b  = (const float*)d_in[7];
    const float* ln_g   = (const float*)d_in[8];
    const float* ln_b   = (const float*)d_in[9];
    const float* lin_W  = (const float*)d_in[10];
    const float* lin_b  = (const float*)d_in[11];
    const float* attr_W = (const float*)d_in[12];
    const float* attr_b = (const float*)d_in[13];
    const float* qkv_W  = (const float*)d_in[14];
    const float* qkv_b  = (const float*)d_in[15];
    const float* out_W  = (const float*)d_in[16];
    const float* out_b  = (const float*)d_in[17];
    const float* fln_g  = (const float*)d_in[18];
    const float* fln_b  = (const float*)d_in[19];
    const float* fuse_W = (const float*)d_in[20];
    const float* fuse_b = (const float*)d_in[21];
    float* out = (float*)d_out;

    const int* src = ei;        // edge_index[0]
    const int* dst = ei + EE;   // edge_index[1]

    // workspace layout (floats); buffers aggressively reused (~256 MB total)
    float* ws = (float*)d_ws;
    const size_t ND = (size_t)NN * DD;
    float* isd  = ws;                 // [N]    degrees -> rsqrt(deg)
    float* csum = isd + NN;           // [128]  column sums
    float* bh   = csum + DD;          // [N,D]  h / aopre0 / preout
    float* bt   = bh + ND;            // [N,D]  hw tmp / attr / aopre1
    float* br   = bt + ND;            // [N,D]  residual
    float* bs   = br + ND;            // [N,D]  struct / ao0
    float* q0b  = bs + ND;            // [N,3D] qkv(struct) / fused-normalized
    float* q1b  = q0b + 3 * ND;       // [N,3D] qkv(attr)   / ao1 (first N*D)

    const dim3 blk(256);
    const int gemmO128  = ((NN / 16) * (DD / 64) + 7) / 8;        // 782 blocks
    const int gemmO384  = ((NN / 16) * (3 * DD / 64) + 7) / 8;    // 2344 blocks
    const int nodeWaves = (NN + 7) / 8;                           // 6250 blocks
    const int edgeWaves = (EE + 7) / 8;                           // 100000 blocks
    const int elemBlk   = ((int)ND + 255) / 256;                  // 25000 blocks

    // --- degrees: deg = 1 + scatter(ones, dst); isd = rsqrt(deg)
    fill_kernel<<<(NN + 255) / 256, blk, 0, stream>>>(isd, 1.0f, NN);
    deg_scatter_kernel<<<(EE + 255) / 256, blk, 0, stream>>>(dst, isd, EE);
    rsqrt_kernel<<<(NN + 255) / 256, blk, 0, stream>>>(isd, NN);

    // --- h = relu(x @ in_W.T + in_b)
    gemm_wmma<true, 1><<<gemmO128, blk, 0, stream>>>(x, in_W, in_b, bh, NN, DD, DD);

    // --- residual = scatter_add((x_org @ res_W)[src], dst)
    gemm_wmma<false, 0><<<gemmO128, blk, 0, stream>>>(x_org, res_W, nullptr, bt, NN, DD, DD);
    fill_kernel<<<elemBlk, blk, 0, stream>>>(br, 0.f, (int)ND);
    res_scatter_kernel<<<edgeWaves, blk, 0, stream>>>(src, dst, bt, br, EE);

    // --- 3x (GCNConv -> LayerNorm -> relu)
    for (int i = 0; i < 3; ++i) {
        gemm_wmma<true, 0><<<gemmO128, blk, 0, stream>>>(
            bh, gcn_W + (size_t)i * DD * DD, nullptr, bt, NN, DD, DD);
        gcn_base_kernel<<<elemBlk, blk, 0, stream>>>(
            bt, isd, gcn_b + (size_t)i * DD, bh, (int)ND);
        gcn_scatter_kernel<<<edgeWaves, blk, 0, stream>>>(src, dst, bt, isd, bh, EE);
        ln128_relu_kernel<<<nodeWaves, blk, 0, stream>>>(
            bh, ln_g + (size_t)i * DD, ln_b + (size_t)i * DD, NN);
    }

    // --- struct = h @ lin_W.T + lin_b ; attr = relu(x_org @ attr_W.T + attr_b)
    gemm_wmma<true, 0><<<gemmO128, blk, 0, stream>>>(bh, lin_W, lin_b, bs, NN, DD, DD);
    gemm_wmma<true, 1><<<gemmO128, blk, 0, stream>>>(x_org, attr_W, attr_b, bt, NN, DD, DD);

    // --- qkv projections (O = 384)
    gemm_wmma<true, 0><<<gemmO384, blk, 0, stream>>>(bs, qkv_W, qkv_b, q0b, NN, DD, 3 * DD);
    gemm_wmma<true, 0><<<gemmO384, blk, 0, stream>>>(bt, qkv_W, qkv_b, q1b, NN, DD, 3 * DD);

    // --- 2x2 attention -> aopre0 (bh), aopre1 (bt)
    attention_kernel<<<nodeWaves, blk, 0, stream>>>(q0b, q1b, bh, bt, NN);

    // --- output projections: ao0 -> bs, ao1 -> q1b (reuse)
    gemm_wmma<true, 0><<<gemmO128, blk, 0, stream>>>(bh, out_W, out_b, bs, NN, DD, DD);
    gemm_wmma<true, 0><<<gemmO128, blk, 0, stream>>>(bt, out_W, out_b, q1b, NN, DD, DD);

    // --- fused = LN384(concat(ao0, ao1, residual)) -> q0b
    fuse_ln384_kernel<<<nodeWaves, blk, 0, stream>>>(bs, q1b, br, fln_g, fln_b, q0b, NN);

    // --- preout = fused @ fuse_W.T + fuse_b (K=384, O=128) -> bh
    gemm_wmma<true, 0><<<gemmO128, blk, 0, stream>>>(q0b, fuse_W, fuse_b, bh, NN, 3 * DD, DD);

    // --- PairNorm PN-SCS (scale 20)
    fill_kernel<<<1, 128, 0, stream>>>(csum, 0.f, DD);
    colsum_kernel<<<256, 128, 0, stream>>>(bh, csum, NN);
    pairnorm_kernel<<<nodeWaves, blk, 0, stream>>>(bh, csum, out, NN);
}